// GCN_27771258536143
// MI455X (gfx1250) — compile-verified
//
#include <hip/hip_runtime.h>

// ---------------------------------------------------------------------------
// GCN forward (3x GCNConv + mean-pool + linear) for MI455X / gfx1250.
// GEMMs use V_WMMA_F32_16X16X4_F32 (fp32, matches reference precision).
// Aggregation is L2-resident atomic scatter (xw table = 12.8MB << 192MB L2).
// ---------------------------------------------------------------------------

typedef float v2f __attribute__((ext_vector_type(2)));
typedef float v8f __attribute__((ext_vector_type(8)));

#define HDIM 64

__global__ void k_fill(float* __restrict__ p, float v, int n) {
  int i = blockIdx.x * blockDim.x + threadIdx.x;
  if (i < n) p[i] = v;
}

// degree accumulation over dst (self-loops pre-seeded as 1.0 by k_fill)
__global__ void k_deg(const int* __restrict__ dst, float* __restrict__ deg, int nE) {
  int e = blockIdx.x * blockDim.x + threadIdx.x;
  if (e < nE) atomicAdd(deg + dst[e], 1.0f);
}

__global__ void k_dinv(const float* __restrict__ deg, float* __restrict__ dinv, int n) {
  int i = blockIdx.x * blockDim.x + threadIdx.x;
  if (i < n) dinv[i] = rsqrtf(deg[i]);   // deg >= 1 always (self-loop)
}

// XW = X @ W, X:[n x 64] row-major, W:[64 x 64] row-major.
// 256 threads = 8 waves per block. Block covers 32 rows x 64 cols.
// Wave w: rows rowBase + (w>>2)*16, cols n0 = (w&3)*16; K loop 64 in steps of 4.
// WMMA f32 16x16x4 operand layouts per CDNA5 ISA 7.12.2.
__global__ __launch_bounds__(256) void k_gemm64(const float* __restrict__ X,
                                                const float* __restrict__ W,
                                                float* __restrict__ XW, int n) {
  __shared__ float sW[HDIM * HDIM];
  for (int i = threadIdx.x; i < HDIM * HDIM; i += 256) sW[i] = W[i];
  __syncthreads();

  const int wave = threadIdx.x >> 5;
  const int lane = threadIdx.x & 31;
  const int rowBase = blockIdx.x * 32 + (wave >> 2) * 16;
  const int n0 = (wave & 3) * 16;
  const int M = lane & 15;            // A row / B,D column index
  const int kOff = (lane >> 4) * 2;   // lanes 0-15 hold K pair {0,1}, 16-31 hold {2,3}

  const int row = rowBase + M;
  const bool inRange = (row < n);
  const int rclamp = inRange ? row : (n - 1);
  const float* xrow = X + (size_t)rclamp * HDIM;

  v8f acc = {0.f, 0.f, 0.f, 0.f, 0.f, 0.f, 0.f, 0.f};
#pragma unroll
  for (int k0 = 0; k0 < HDIM; k0 += 4) {
    v2f a;                                     // A 16x4: lane M, K = k0+kOff(+1)
    a.x = xrow[k0 + kOff];
    a.y = xrow[k0 + kOff + 1];
    if (!inRange) { a.x = 0.f; a.y = 0.f; }
    v2f b;                                     // B 4x16: K = k0+kOff(+1), N = n0+M
    b.x = sW[(k0 + kOff) * HDIM + n0 + M];
    b.y = sW[(k0 + kOff + 1) * HDIM + n0 + M];
    // (neg_a, A, neg_b, B, c_mod, C, reuse_a, reuse_b)
    acc = __builtin_amdgcn_wmma_f32_16x16x4_f32(false, a, false, b,
                                                (short)0, acc, false, false);
  }

  // D 16x16: element v -> row rowBase + v + 8*(lane>>4), col n0 + (lane&15)
  const int dRow0 = rowBase + ((lane >> 4) << 3);
  const int col = n0 + M;
#pragma unroll
  for (int v = 0; v < 8; ++v) {
    int r = dRow0 + v;
    if (r < n) XW[(size_t)r * HDIM + col] = acc[v];
  }
}

// agg[i][:] = dinv[i]^2 * xw[i][:]   (self-loop contribution; non-atomic init)
__global__ void k_selfloop(const float* __restrict__ xw, const float* __restrict__ dinv,
                           float* __restrict__ agg, int n) {
  int t = blockIdx.x * blockDim.x + threadIdx.x;
  int i = t >> 6;
  if (i < n) {
    float w = dinv[i] * dinv[i];
    agg[t] = w * xw[t];
  }
}

// per-edge scatter: agg[dst] += dinv[src]*dinv[dst] * xw[src]
// 16 threads per edge, float4 per thread (gathers hit L2).
__global__ void k_scatter(const int* __restrict__ src, const int* __restrict__ dst,
                          const float* __restrict__ dinv, const float* __restrict__ xw,
                          float* __restrict__ agg, int nE) {
  int t = blockIdx.x * blockDim.x + threadIdx.x;
  int e = t >> 4;
  if (e >= nE) return;
  int f4 = (t & 15) * 4;
  int s = src[e];
  int d = dst[e];
  float norm = dinv[s] * dinv[d];
  const float4 xv = *(const float4*)(xw + (size_t)s * HDIM + f4);
  float* dp = agg + (size_t)d * HDIM + f4;
  atomicAdd(dp + 0, norm * xv.x);
  atomicAdd(dp + 1, norm * xv.y);
  atomicAdd(dp + 2, norm * xv.z);
  atomicAdd(dp + 3, norm * xv.w);
}

// h = (agg + bias), optional ReLU
__global__ void k_bias_act(const float* __restrict__ agg, const float* __restrict__ b,
                           float* __restrict__ out, int n, int relu) {
  int t = blockIdx.x * blockDim.x + threadIdx.x;
  int i = t >> 6;
  int h = t & 63;
  if (i < n) {
    float v = agg[t] + b[h];
    out[t] = relu ? fmaxf(v, 0.f) : v;
  }
}

// global mean pool: sums + counts via atomics
__global__ void k_pool(const float* __restrict__ h3, const int* __restrict__ batch,
                       float* __restrict__ pooled, float* __restrict__ cnt, int n) {
  int t = blockIdx.x * blockDim.x + threadIdx.x;
  int i = t >> 6;
  int h = t & 63;
  if (i < n) {
    int g = batch[i];
    atomicAdd(pooled + (size_t)g * HDIM + h, h3[t]);
    if (h == 0) atomicAdd(cnt + g, 1.0f);
  }
}

// out[g][o] = blin[o] + sum_h pooled[g][h]/max(cnt,1) * Wlin[h][o]
__global__ void k_final(const float* __restrict__ pooled, const float* __restrict__ cnt,
                        const float* __restrict__ Wlin, const float* __restrict__ blin,
                        float* __restrict__ out, int nG) {
  int t = blockIdx.x * blockDim.x + threadIdx.x;
  if (t < nG * 2) {
    int g = t >> 1;
    int o = t & 1;
    float inv = 1.0f / fmaxf(cnt[g], 1.0f);
    float s = blin[o];
#pragma unroll
    for (int h = 0; h < HDIM; ++h)
      s += pooled[(size_t)g * HDIM + h] * inv * Wlin[h * 2 + o];
    out[t] = s;
  }
}

extern "C" void kernel_launch(void* const* d_in, const int* in_sizes, int n_in,
                              void* d_out, int out_size, void* d_ws, size_t ws_size,
                              hipStream_t stream) {
  const float* x    = (const float*)d_in[0];
  const float* W1   = (const float*)d_in[1];
  const float* b1   = (const float*)d_in[2];
  const float* W2   = (const float*)d_in[3];
  const float* b2   = (const float*)d_in[4];
  const float* W3   = (const float*)d_in[5];
  const float* b3   = (const float*)d_in[6];
  const float* Wlin = (const float*)d_in[7];
  const float* blin = (const float*)d_in[8];
  const int* edge   = (const int*)d_in[9];   // [2, nE] flattened
  const int* batch  = (const int*)d_in[10];

  const int n  = in_sizes[0] / HDIM;   // 50000
  const int nE = in_sizes[9] / 2;      // 800000
  const int nG = out_size / 2;         // 512
  const int* src = edge;
  const int* dst = edge + nE;

  float* ws = (float*)d_ws;
  size_t off = 0;
  float* xw     = ws + off; off += (size_t)n * HDIM;
  float* agg    = ws + off; off += (size_t)n * HDIM;
  float* hbuf   = ws + off; off += (size_t)n * HDIM;
  float* deg    = ws + off; off += n;
  float* dinv   = ws + off; off += n;
  float* pooled = ws + off; off += (size_t)nG * HDIM;
  float* cnt    = ws + off; off += nG;

  auto cdiv = [](int a, int b) { return (a + b - 1) / b; };
  const dim3 B(256);

  // degree / normalization (deterministic: re-initialized every call)
  k_fill<<<cdiv(n, 256), B, 0, stream>>>(deg, 1.0f, n);
  k_deg<<<cdiv(nE, 256), B, 0, stream>>>(dst, deg, nE);
  k_dinv<<<cdiv(n, 256), B, 0, stream>>>(deg, dinv, n);

  const float* Ws[3] = {W1, W2, W3};
  const float* bs[3] = {b1, b2, b3};
  const float* cur = x;
  for (int l = 0; l < 3; ++l) {
    k_gemm64<<<cdiv(n, 32), B, 0, stream>>>(cur, Ws[l], xw, n);
    k_selfloop<<<cdiv(n * HDIM, 256), B, 0, stream>>>(xw, dinv, agg, n);
    k_scatter<<<cdiv(nE * 16, 256), B, 0, stream>>>(src, dst, dinv, xw, agg, nE);
    k_bias_act<<<cdiv(n * HDIM, 256), B, 0, stream>>>(agg, bs[l], hbuf, n, (l < 2) ? 1 : 0);
    cur = hbuf;
  }

  k_fill<<<cdiv(nG * HDIM, 256), B, 0, stream>>>(pooled, 0.0f, nG * HDIM);
  k_fill<<<cdiv(nG, 256), B, 0, stream>>>(cnt, 0.0f, nG);
  k_pool<<<cdiv(n * HDIM, 256), B, 0, stream>>>(hbuf, batch, pooled, cnt, n);
  k_final<<<cdiv(nG * 2, 256), B, 0, stream>>>(pooled, cnt, Wlin, blin, (float*)d_out, nG);
}